// CD_88364657148397
// MI455X (gfx1250) — compile-verified
//
#include <hip/hip_runtime.h>
#include <math.h>

typedef float v2f __attribute__((ext_vector_type(2)));
typedef float v8f __attribute__((ext_vector_type(8)));

#define WAVES_PER_BLOCK 8
#define ROWS_PER_BLOCK  (WAVES_PER_BLOCK * 16)
#define INF_BITS 0x7F800000u

// ---------------------------------------------------------------------------
// Init workspace: colmin[B*M] = +INF (bits), rowsum[B] = 0
// ---------------------------------------------------------------------------
__global__ void chamfer_init(unsigned int* __restrict__ colmin,
                             float* __restrict__ rowsum,
                             int total_cols, int B) {
    int i = blockIdx.x * blockDim.x + threadIdx.x;
    if (i < total_cols) colmin[i] = INF_BITS;
    if (i < B) rowsum[i] = 0.0f;
}

// ---------------------------------------------------------------------------
// Main kernel: each wave owns 16 rows of pc1, sweeps all M cols of pc2 in
// 16-wide tiles, TWO tiles per loop iteration (two independent WMMA streams
// so each WMMA's VALU consumers overlap the other WMMA's execution).
//
// Dot products via V_WMMA_F32_16X16X4_F32 (K padded 3 -> 4).
// d2 = |a|^2 + |b|^2 - 2 a.b, clamped to >= 0.
// Boundary handling folded into the norms (invalid row/col -> +INF term):
// the inner loop is completely branchless; +INF propagates through fma/fmax
// and is neutral for both min reductions.
// ---------------------------------------------------------------------------
__global__ __launch_bounds__(256)
void chamfer_wmma(const float* __restrict__ pc1,     // [B, N, 3]
                  const float* __restrict__ pc2,     // [B, M, 3]
                  float* __restrict__ rowsum,        // [B]
                  unsigned int* __restrict__ colmin, // [B, M]
                  int N, int M) {
    const int b      = blockIdx.y;
    const int waveId = threadIdx.x >> 5;
    const int lane   = threadIdx.x & 31;
    const int half   = lane >> 4;       // 0: lanes 0-15, 1: lanes 16-31
    const int l16    = lane & 15;

    const int rowbase = blockIdx.x * ROWS_PER_BLOCK + waveId * 16;

    // ---- A fragment: 16x4 f32, K = {x, y, z, 0} ----------------------------
    // ISA layout: lanes 0-15 -> M=lane,    VGPR0=K0, VGPR1=K1
    //             lanes16-31 -> M=lane-16, VGPR0=K2, VGPR1=K3
    const int arow   = rowbase + l16;
    const int arow_c = (arow < N) ? arow : (N - 1);
    const float* ap  = pc1 + ((size_t)b * N + arow_c) * 3;
    const float ax = ap[0], ay = ap[1], az = ap[2];
    const float a2 = fmaf(ax, ax, fmaf(ay, ay, az * az));

    v2f afrag;
    afrag.x = half ? az   : ax;   // K2 : K0
    afrag.y = half ? 0.0f : ay;   // K3 : K1

    // |a|^2 for the rows this lane sees in the D fragment; invalid row -> +INF.
    float a2eff[8];
    #pragma unroll
    for (int r = 0; r < 8; ++r) {
        const int rr = half ? (r + 8) : r;
        const float v = __shfl(a2, rr, 32);
        a2eff[r] = (rowbase + rr < N) ? v : INFINITY;
    }

    float rowmin[8];
    #pragma unroll
    for (int r = 0; r < 8; ++r) rowmin[r] = INFINITY;

    const float* bbat = pc2 + (size_t)b * M * 3;
    unsigned int* cmbat = colmin + (size_t)b * M;

    const int T  = (M + 15) >> 4;   // 16-col tiles
    const int TP = (T + 1) >> 1;    // tile pairs

    // Clamped tile loader: always in-bounds; validity returned separately.
    auto loadTile = [&](int t, float& x, float& y, float& z, bool& okk) {
        const int col = t * 16 + l16;
        const int cc  = (col < M) ? col : (M - 1);
        const float* p = bbat + (size_t)cc * 3;
        x = p[0]; y = p[1]; z = p[2];
        okk = (col < M);
    };

    // One tile's WMMA + combine + dual-min + column atomic.
    auto processTile = [&](int t, float bx, float by, float bz, bool ok) {
        const float b2 = fmaf(bx, bx, fmaf(by, by, bz * bz));
        const float b2eff = ok ? b2 : INFINITY;

        v2f bfrag;                       // B: 4x16 f32 (KxN), mirror of A
        bfrag.x = half ? bz   : bx;      // K2 : K0
        bfrag.y = half ? 0.0f : by;      // K3 : K1

        v8f acc = {};
        acc = __builtin_amdgcn_wmma_f32_16x16x4_f32(
            false, afrag, false, bfrag, (short)0, acc, false, false);

        float cmin = INFINITY;
        #pragma unroll
        for (int r = 0; r < 8; ++r) {
            float df = fmaf(-2.0f, acc[r], a2eff[r] + b2eff);
            df = fmaxf(df, 0.0f);
            rowmin[r] = fminf(rowmin[r], df);
            cmin = fminf(cmin, df);
        }
        if (ok)
            atomicMin(&cmbat[t * 16 + l16], __float_as_uint(cmin));
    };

    // ---- Software-pipelined sweep, two independent tiles per iteration -----
    float bx0, by0, bz0, bx1, by1, bz1;
    bool  ok0, ok1;
    loadTile(0, bx0, by0, bz0, ok0);
    loadTile(1, bx1, by1, bz1, ok1);

    for (int i = 0; i < TP; ++i) {
        const int t0 = 2 * i;

        // Issue next pair's loads early: they overlap this pair's WMMA+min.
        float nx0, ny0, nz0, nx1, ny1, nz1;
        bool  nok0, nok1;
        loadTile(t0 + 2, nx0, ny0, nz0, nok0);
        loadTile(t0 + 3, nx1, ny1, nz1, nok1);

        // Prefetch four tiles ahead (gfx1250 global_prefetch_b8).
        {
            const int pcol = (t0 + 4) * 16 + l16;
            const int pcc  = (pcol < M) ? pcol : (M - 1);
            __builtin_prefetch(bbat + (size_t)pcc * 3, 0, 1);
        }

        processTile(t0,     bx0, by0, bz0, ok0);
        processTile(t0 + 1, bx1, by1, bz1, ok1);

        bx0 = nx0; by0 = ny0; bz0 = nz0; ok0 = nok0;
        bx1 = nx1; by1 = ny1; bz1 = nz1; ok1 = nok1;
    }

    // ---- Row-min reduction across the 16-lane half groups ------------------
    float s = 0.0f;
    #pragma unroll
    for (int r = 0; r < 8; ++r) {
        float v = rowmin[r];
        v = fminf(v, __shfl_xor(v, 8, 16));
        v = fminf(v, __shfl_xor(v, 4, 16));
        v = fminf(v, __shfl_xor(v, 2, 16));
        v = fminf(v, __shfl_xor(v, 1, 16));
        const int rrow = rowbase + (half ? (r + 8) : r);
        if (l16 == 0 && rrow < N) s += v;
    }
    if (l16 == 0)
        atomicAdd(&rowsum[b], s);   // lane 0: rows +0..7, lane 16: rows +8..15
}

// ---------------------------------------------------------------------------
// Final reduction: one block per batch sums colmin over M.
// out[b] = rowsum[b]/N + colsum/M
// ---------------------------------------------------------------------------
__global__ __launch_bounds__(256)
void chamfer_reduce(const unsigned int* __restrict__ colmin,
                    const float* __restrict__ rowsum,
                    float* __restrict__ out, int N, int M) {
    const int b = blockIdx.x;
    __shared__ float smem[256];
    float s = 0.0f;
    for (int i = threadIdx.x; i < M; i += 256)
        s += __uint_as_float(colmin[(size_t)b * M + i]);
    smem[threadIdx.x] = s;
    __syncthreads();
    for (int off = 128; off > 0; off >>= 1) {
        if (threadIdx.x < off) smem[threadIdx.x] += smem[threadIdx.x + off];
        __syncthreads();
    }
    if (threadIdx.x == 0)
        out[b] = rowsum[b] / (float)N + smem[0] / (float)M;
}

// ---------------------------------------------------------------------------
extern "C" void kernel_launch(void* const* d_in, const int* in_sizes, int n_in,
                              void* d_out, int out_size, void* d_ws, size_t ws_size,
                              hipStream_t stream) {
    const float* pc1 = (const float*)d_in[0];   // [B, N, 3] f32
    const float* pc2 = (const float*)d_in[1];   // [B, M, 3] f32
    float* out = (float*)d_out;                 // [B] f32

    const int B = out_size;                     // reference returns [B]
    const int N = in_sizes[0] / (B * 3);
    const int M = in_sizes[1] / (B * 3);

    unsigned int* colmin = (unsigned int*)d_ws;            // B*M uints
    float* rowsum = (float*)((char*)d_ws + (size_t)B * M * sizeof(unsigned int)); // B floats

    // 1) init workspace (must run every launch: graph replay, poisoned ws)
    {
        const int total = B * M;
        const int blocks = (total + 255) / 256;
        chamfer_init<<<blocks, 256, 0, stream>>>(colmin, rowsum, total, B);
    }

    // 2) main tiled WMMA kernel
    {
        dim3 grid((N + ROWS_PER_BLOCK - 1) / ROWS_PER_BLOCK, B);
        chamfer_wmma<<<grid, 256, 0, stream>>>(pc1, pc2, rowsum, colmin, N, M);
    }

    // 3) final reduction -> out[b]
    chamfer_reduce<<<B, 256, 0, stream>>>(colmin, rowsum, out, N, M);
}